// RegionLoss_2353642078367
// MI455X (gfx1250) — compile-verified
//
#include <hip/hip_runtime.h>
#include <math.h>

typedef float v2f __attribute__((ext_vector_type(2)));
typedef float v8f __attribute__((ext_vector_type(8)));

#define NB 128
#define NT 30
#define NG 13
#define NCLS 16
#define NCELL 169
#define ROWS 507                 // 169*3 rows actually used
#define CH 21                    // 5 + NCLS channels
#define BATCH_STRIDE (10647*21)  // full output row count per batch * 21
#define WNORM 416.0f
#define OBJ_SQRT5 2.2360679774997896f

__global__ void rl_zero(float* out) { out[0] = 0.0f; }

// Pass 1: baseline conf loss  0.5 * sum_{b,i,a} m[b,i,a,4]^2
// One conf value per thread (no loop -> no serialized load chain; 2032 full
// waves hide all latency). Wave reduction via V_WMMA_F32_16X16X4_F32 against
// an all-ones B matrix, then block reduction through LDS (ds_add_f32), then
// one global atomic per block.
__global__ __launch_bounds__(256) void rl_conf_base(const float* __restrict__ outp,
                                                    float* __restrict__ loss) {
    const int e = blockIdx.x * 256 + threadIdx.x;
    const int N = NB * ROWS;

    float v = 0.0f;
    if (e < N) {
        int b = e / ROWS;
        int j = e - b * ROWS;
        v = outp[(size_t)b * BATCH_STRIDE + (size_t)j * CH + 4];
    }
    float s = v * v;   // EXEC re-converged here; tail lanes carry 0

    // A (16x4 f32): lanes 0-15 hold K=0,1 ; lanes 16-31 hold K=2,3.
    // With B = ones(4x16): D[m,n] = rowsum(m) replicated across columns.
    v2f a;     a.x = s;    a.y = 0.0f;
    v2f bones; bones.x = 1.0f; bones.y = 1.0f;
    v8f c = {};
    v8f d = __builtin_amdgcn_wmma_f32_16x16x4_f32(false, a, false, bones,
                                                  (short)0, c, false, false);
    float t = d[0] + d[1] + d[2] + d[3] + d[4] + d[5] + d[6] + d[7];

    // lane 0 holds sum over M=0..7, lane 16 holds sum over M=8..15
    __shared__ float bsum;
    if (threadIdx.x == 0) bsum = 0.0f;
    __syncthreads();
    int lane = threadIdx.x & 31;
    if (lane == 0 || lane == 16) atomicAdd(&bsum, t);
    __syncthreads();
    if (threadIdx.x == 0) atomicAdd(loss, 0.5f * bsum);
}

// Pass 2: one thread per (batch, target). Computes IoUs vs 3 anchors,
// argmax anchor, conf-loss correction at the target cell, plus coord and
// class losses at the max anchor. (Input generator guarantees distinct
// cells per batch, so deltas are exact.)
__global__ __launch_bounds__(128) void rl_targets(const float* __restrict__ outp,
                                                  const float* __restrict__ tgt,
                                                  float* __restrict__ loss) {
    int gid = blockIdx.x * blockDim.x + threadIdx.x;   // 0 .. NB*NT-1 exactly
    int b = gid / NT;
    int t = gid - b * NT;

    const float* tg = tgt + (size_t)(b * NT + t) * 5;
    float x1 = tg[0], y1 = tg[1], x2 = tg[2], y2 = tg[3];
    int   cls = (int)tg[4];

    float lx = 0.5f * (x1 + x2);
    float ly = 0.5f * (y1 + y2);
    float tw = x2 - x1;
    float th = y2 - y1;

    int gx = (int)floorf(lx * 0.03125f);   // /32 (exact, power of two)
    int gy = (int)floorf(ly * 0.03125f);
    int idx = gy * NG + gx;
    if (idx < 0) idx = 0;
    if (idx >= NCELL) idx = NCELL - 1;     // matches jax clamped gather

    const float* mb = outp + (size_t)b * BATCH_STRIDE + (size_t)idx * 3 * CH;

    float m0[3], m1[3], m2[3], m3[3], m4[3], iou[3];
    float ta = tw * th;
#pragma unroll
    for (int a = 0; a < 3; ++a) {
        const float* mp = mb + a * CH;
        m0[a] = mp[0]; m1[a] = mp[1]; m2[a] = mp[2]; m3[a] = mp[3]; m4[a] = mp[4];
        float px1 = m0[a] - 0.5f * m2[a];
        float px2 = m0[a] + 0.5f * m2[a];
        float py1 = m1[a] - 0.5f * m3[a];
        float py2 = m1[a] + 0.5f * m3[a];
        float ix1 = fmaxf(px1, x1), iy1 = fmaxf(py1, y1);
        float ix2 = fminf(px2, x2), iy2 = fminf(py2, y2);
        float inter = fmaxf(ix2 - ix1, 0.0f) * fmaxf(iy2 - iy1, 0.0f);
        float pa = (px2 - px1) * (py2 - py1);
        iou[a] = inter / (pa + ta - inter + 1e-16f);
    }

    // first-max argmax (jnp.argmax semantics)
    int amax = 0;
    float miou = iou[0];
    if (iou[1] > miou) { miou = iou[1]; amax = 1; }
    if (iou[2] > miou) { miou = iou[2]; amax = 2; }

    float acc = 0.0f;

    // conf-loss delta: replace baseline m4^2 with masked term
#pragma unroll
    for (int a = 0; a < 3; ++a) {
        float cv, tc;
        if (a == amax) { cv = OBJ_SQRT5; tc = miou; }
        else           { cv = (iou[a] > 0.5f) ? 0.0f : 1.0f; tc = 0.0f; }
        float u = cv * (m4[a] - tc);
        acc += u * u - m4[a] * m4[a];
    }

    // coord loss at max anchor
    float wgt = 2.0f - (tw / WNORM) * (th / WNORM);
    float dx = wgt * (m0[amax] / WNORM) - wgt * (lx / WNORM);
    float dy = wgt * (m1[amax] / WNORM) - wgt * (ly / WNORM);
    float dw = wgt * (m2[amax] / WNORM) - wgt * (tw / WNORM);
    float dh = wgt * (m3[amax] / WNORM) - wgt * (th / WNORM);
    acc += dx * dx + dy * dy + dw * dw + dh * dh;

    // class loss at max anchor
    const float* mc = mb + amax * CH + 5;
#pragma unroll
    for (int c = 0; c < NCLS; ++c) {
        float e = mc[c] - ((c == cls) ? 1.0f : 0.0f);
        acc += e * e;
    }

    atomicAdd(loss, 0.5f * acc);
}

extern "C" void kernel_launch(void* const* d_in, const int* in_sizes, int n_in,
                              void* d_out, int out_size, void* d_ws, size_t ws_size,
                              hipStream_t stream) {
    const float* outp = (const float*)d_in[0];   // (128, 10647, 21) fp32
    const float* tgt  = (const float*)d_in[1];   // (128, 30, 5)     fp32
    float* loss = (float*)d_out;                 // scalar fp32

    const int N = NB * ROWS;
    rl_zero<<<1, 1, 0, stream>>>(loss);
    rl_conf_base<<<(N + 255) / 256, 256, 0, stream>>>(outp, loss);
    rl_targets<<<NB * NT / 128, 128, 0, stream>>>(outp, tgt, loss);
}